// VGAEEncoder_43989055045969
// MI455X (gfx1250) — compile-verified
//
#include <hip/hip_runtime.h>
#include <hip/hip_bf16.h>
#include <math.h>

typedef float v2f __attribute__((ext_vector_type(2)));
typedef float v8f __attribute__((ext_vector_type(8)));

#define NEG_SLOPE 0.2f

// ---------------- helpers ----------------

// Monotone float -> uint key so atomicMax(unsigned) == float max (handles negatives).
static __device__ __forceinline__ unsigned f2key(float f) {
  int b = __float_as_int(f);
  return (b >= 0) ? ((unsigned)b | 0x80000000u) : ~(unsigned)b;
}
static __device__ __forceinline__ float key2f(unsigned u) {
  int b = (u & 0x80000000u) ? (int)(u & 0x7fffffffu) : (int)(~u);
  return __int_as_float(b);
}

// host-side copy of f2key(-inf): bits(-inf)=0xFF800000 (negative) -> ~bits = 0x007FFFFF
static inline unsigned f2key_host() { return 0x007FFFFFu; }

// edge e in [0,E): from edge_index (int64, [2,E] flat). e in [E, E+N): self loop.
static __device__ __forceinline__ void edge_sd(const long long* __restrict__ ei, int E, int e,
                                               int& s, int& d) {
  if (e < E) { s = (int)ei[e]; d = (int)ei[(size_t)E + e]; }
  else       { s = e - E; d = e - E; }
}

// ---------------- FP32 WMMA GEMM ----------------
// C[M,N] = A[M,K] @ B[K,N], f32. Requires M%16==0, N%(16*NT)==0, K%16==0.
// One wave32 computes a 16 x (16*NT) strip using V_WMMA_F32_16X16X4_F32.
// K is processed in blocks of 16 (4 WMMA k-steps): all fragments for a block are
// loaded into registers first so the loads clause together and the 4*NT WMMAs
// issue back-to-back instead of one s_wait_loadcnt per WMMA.
template<int NT>
__global__ __launch_bounds__(256)
void gemm_wmma_f32(const float* __restrict__ A, const float* __restrict__ B,
                   float* __restrict__ C, int M, int K, int N)
{
  const int wave = blockIdx.x * 8 + (threadIdx.x >> 5);   // 256 threads = 8 waves
  const int lane = threadIdx.x & 31;
  const int strips = N / (16 * NT);
  const int mtiles = M >> 4;
  if (wave >= mtiles * strips) return;   // wave-uniform exit; EXEC all-ones at WMMA

  const int tileM = wave / strips;
  const int strip = wave % strips;
  const int m0 = tileM << 4;
  const int n0 = strip * 16 * NT;
  const int mrow  = lane & 15;
  const int khalf = (lane >> 4) << 1;    // 0 or 2

  v8f acc[NT] = {};
  const float* __restrict__ Arow = A + (size_t)(m0 + mrow) * K;

  for (int k0 = 0; k0 < K; k0 += 16) {
    v2f a[4];
    v2f b[4][NT];
#pragma unroll
    for (int s = 0; s < 4; ++s) {
      const int kk = k0 + 4 * s + khalf;
      a[s].x = Arow[kk];
      a[s].y = Arow[kk + 1];
      const float* __restrict__ Bp = B + (size_t)kk * N + n0 + mrow;
#pragma unroll
      for (int t = 0; t < NT; ++t) {
        b[s][t].x = Bp[t * 16];
        b[s][t].y = Bp[t * 16 + N];
      }
    }
#pragma unroll
    for (int s = 0; s < 4; ++s)
#pragma unroll
      for (int t = 0; t < NT; ++t)
        acc[t] = __builtin_amdgcn_wmma_f32_16x16x4_f32(
            false, a[s], false, b[s][t], (short)0, acc[t], false, false);
  }

  const int rowoff = (lane >> 4) * 8;
#pragma unroll
  for (int t = 0; t < NT; ++t) {
    float* __restrict__ Cp = C + (size_t)(m0 + rowoff) * N + n0 + t * 16 + mrow;
#pragma unroll
    for (int r = 0; r < 8; ++r) Cp[(size_t)r * N] = acc[t][r];
  }
}

// ---------------- elementwise / edge kernels ----------------

__global__ void fill_f32(float* __restrict__ p, float v, long long n) {
  long long i = (long long)blockIdx.x * blockDim.x + threadIdx.x;
  if (i < n) p[i] = v;
}
__global__ void fill_u32(unsigned* __restrict__ p, unsigned v, long long n) {
  long long i = (long long)blockIdx.x * blockDim.x + threadIdx.x;
  if (i < n) p[i] = v;
}

// s[n,h] = <xl[n,h,:], a_src[h,:]> ; d likewise
__global__ void attn_scores(const float* __restrict__ xl,
                            const float* __restrict__ asr, const float* __restrict__ adt,
                            float* __restrict__ s, float* __restrict__ d,
                            int N, int H, int C)
{
  int idx = blockIdx.x * blockDim.x + threadIdx.x;
  if (idx >= N * H) return;
  int n = idx / H, h = idx % H;
  const float* xp = xl + (size_t)n * H * C + (size_t)h * C;
  const float* ap = asr + (size_t)h * C;
  const float* bp = adt + (size_t)h * C;
  float ss = 0.f, dd = 0.f;
  for (int c = 0; c < C; ++c) { float v = xp[c]; ss += v * ap[c]; dd += v * bp[c]; }
  s[idx] = ss; d[idx] = dd;
}

// logit = leaky_relu(s[src]+d[dst]); store; atomicMax key into mkeys[dst]
__global__ void edge_logits_max(const long long* __restrict__ ei, int E, int Etot, int H,
                                const float* __restrict__ s, const float* __restrict__ d,
                                float* __restrict__ logits, unsigned* __restrict__ mkeys)
{
  long long idx = (long long)blockIdx.x * blockDim.x + threadIdx.x;
  if (idx >= (long long)Etot * H) return;
  int e = (int)(idx / H), h = (int)(idx % H);
  int sj, di; edge_sd(ei, E, e, sj, di);
  float lg = s[(size_t)sj * H + h] + d[(size_t)di * H + h];
  lg = (lg > 0.f) ? lg : NEG_SLOPE * lg;
  logits[idx] = lg;
  atomicMax(&mkeys[(size_t)di * H + h], f2key(lg));
}

// e = exp(logit - m[dst]); overwrite logits with e; atomicAdd denom[dst]
__global__ void edge_exp_sum(const long long* __restrict__ ei, int E, int Etot, int H,
                             const unsigned* __restrict__ mkeys,
                             float* __restrict__ logits, float* __restrict__ denom)
{
  long long idx = (long long)blockIdx.x * blockDim.x + threadIdx.x;
  if (idx >= (long long)Etot * H) return;
  int e = (int)(idx / H), h = (int)(idx % H);
  int sj, di; edge_sd(ei, E, e, sj, di);
  float m = key2f(mkeys[(size_t)di * H + h]);
  float ev = expf(logits[idx] - m);
  logits[idx] = ev;
  atomicAdd(&denom[(size_t)di * H + h], ev);
}

// out[dst, f..f+3] += (e[e,h]/denom[dst,h]) * xl[src, f..f+3]   (alpha fused in)
__global__ void edge_aggregate4(const long long* __restrict__ ei, int E, int Etot, int H, int C,
                                const float* __restrict__ enum_, const float* __restrict__ denom,
                                const float* __restrict__ xl, float* __restrict__ out)
{
  const int F  = H * C;
  const int F4 = F >> 2;
  long long idx = (long long)blockIdx.x * blockDim.x + threadIdx.x;
  long long total = (long long)Etot * F4;
  if (idx >= total) return;
  int e = (int)(idx / F4), q = (int)(idx % F4);
  int f = q << 2;
  int h = f / C;
  int sj, di; edge_sd(ei, E, e, sj, di);
  float a = enum_[(size_t)e * H + h] / denom[(size_t)di * H + h];
  const float4 xv = *(const float4*)(xl + (size_t)sj * F + f);
  float* op = out + (size_t)di * F + f;
  atomicAdd(op + 0, a * xv.x);
  atomicAdd(op + 1, a * xv.y);
  atomicAdd(op + 2, a * xv.z);
  atomicAdd(op + 3, a * xv.w);
}

// x = act(x + bias[f]); act: 1 = ELU, 0 = identity
__global__ void bias_act(float* __restrict__ x, const float* __restrict__ b,
                         int N, int F, int act)
{
  long long idx = (long long)blockIdx.x * blockDim.x + threadIdx.x;
  if (idx >= (long long)N * F) return;
  int f = (int)(idx % F);
  float v = x[idx] + b[f];
  if (act) v = (v > 0.f) ? v : expm1f(v);
  x[idx] = v;
}

// ---------------- host orchestration ----------------

static inline int cdiv_ll(long long a, int b) { return (int)((a + b - 1) / b); }

static void run_gat(const float* X, int Fin, const float* W,
                    const float* asr, const float* adt, const float* bias,
                    int H, int C, const long long* ei, int E, int N,
                    float* xl, float* sbuf, float* dbuf, unsigned* mkeys, float* denom,
                    float* logits, float* out, int act, hipStream_t stream)
{
  const int F = H * C;
  const int Etot = E + N;

  // 1) xl = X @ W  (FP32 WMMA)
  if (F % 64 == 0) {
    int waves = (N / 16) * (F / 64);
    gemm_wmma_f32<4><<<(waves + 7) / 8, 256, 0, stream>>>(X, W, xl, N, Fin, F);
  } else {
    int waves = (N / 16) * (F / 32);
    gemm_wmma_f32<2><<<(waves + 7) / 8, 256, 0, stream>>>(X, W, xl, N, Fin, F);
  }

  // 2) per-node attention scores
  attn_scores<<<cdiv_ll((long long)N * H, 256), 256, 0, stream>>>(xl, asr, adt, sbuf, dbuf, N, H, C);

  // 3) init segment-max keys / denom / accumulator
  fill_u32<<<cdiv_ll((long long)N * H, 256), 256, 0, stream>>>(mkeys, f2key_host(), (long long)N * H);
  fill_f32<<<cdiv_ll((long long)N * H, 256), 256, 0, stream>>>(denom, 0.f, (long long)N * H);
  fill_f32<<<cdiv_ll((long long)N * F, 256), 256, 0, stream>>>(out, 0.f, (long long)N * F);

  // 4) edge logits + segment max
  edge_logits_max<<<cdiv_ll((long long)Etot * H, 256), 256, 0, stream>>>(ei, E, Etot, H, sbuf, dbuf, logits, mkeys);
  // 5) exp + segment sum
  edge_exp_sum<<<cdiv_ll((long long)Etot * H, 256), 256, 0, stream>>>(ei, E, Etot, H, mkeys, logits, denom);
  // 6) weighted aggregation (alpha = e/denom fused; float4 per thread)
  edge_aggregate4<<<cdiv_ll((long long)Etot * (F >> 2), 256), 256, 0, stream>>>(ei, E, Etot, H, C, logits, denom, xl, out);
  // 7) bias + activation
  bias_act<<<cdiv_ll((long long)N * F, 256), 256, 0, stream>>>(out, bias, N, F, act);
}

extern "C" void kernel_launch(void* const* d_in, const int* in_sizes, int n_in,
                              void* d_out, int out_size, void* d_ws, size_t ws_size,
                              hipStream_t stream)
{
  const float*     x   = (const float*)d_in[0];
  const long long* ei  = (const long long*)d_in[1];
  const float*     W1  = (const float*)d_in[2];
  const float*     as1 = (const float*)d_in[3];
  const float*     ad1 = (const float*)d_in[4];
  const float*     b1  = (const float*)d_in[5];
  const float*     W2  = (const float*)d_in[6];
  const float*     as2 = (const float*)d_in[7];
  const float*     ad2 = (const float*)d_in[8];
  const float*     b2  = (const float*)d_in[9];
  const float*     Wmu = (const float*)d_in[10];
  const float*     asmu= (const float*)d_in[11];
  const float*     admu= (const float*)d_in[12];
  const float*     bmu = (const float*)d_in[13];
  const float*     Wls = (const float*)d_in[14];
  const float*     asls= (const float*)d_in[15];
  const float*     adls= (const float*)d_in[16];
  const float*     bls = (const float*)d_in[17];

  const int Fin = 128;
  const int N   = in_sizes[0] / Fin;   // 30000 (multiple of 16)
  const int E   = in_sizes[1] / 2;     // 480000
  const int H1 = 8, C1 = 64, Hid = 64, Out = 32;
  const int Etot = E + N;

  // workspace layout (f32 elements; every buffer offset is a multiple of 16 B)
  float* p = (float*)d_ws;
  float* xl     = p; p += (size_t)N * (H1 * C1);   // projected features (max size, reused)
  float* h1     = p; p += (size_t)N * (H1 * C1);   // layer-1 output
  float* h2     = p; p += (size_t)N * Hid;         // layer-2 output
  float* logits = p; p += (size_t)Etot * H1;       // per-edge logits/e (max size, reused)
  float* sbuf   = p; p += (size_t)N * H1;
  float* dbuf   = p; p += (size_t)N * H1;
  float* denom  = p; p += (size_t)N * H1;
  unsigned* mkeys = (unsigned*)p;                   // N*H1 u32

  float* mu = (float*)d_out;
  float* ls = (float*)d_out + (size_t)N * Out;

  // GAT-1: 8 heads x 64, concat, ELU
  run_gat(x,  Fin,     W1,  as1,  ad1,  b1,  H1, C1, ei, E, N, xl, sbuf, dbuf, mkeys, denom, logits, h1, 1, stream);
  // GAT-2: 1 head x 64, mean(=identity), ELU
  run_gat(h1, H1 * C1, W2,  as2,  ad2,  b2,  1,  Hid, ei, E, N, xl, sbuf, dbuf, mkeys, denom, logits, h2, 1, stream);
  // mu / logstd heads: 1 head x 32, no activation
  run_gat(h2, Hid,     Wmu, asmu, admu, bmu, 1,  Out, ei, E, N, xl, sbuf, dbuf, mkeys, denom, logits, mu, 0, stream);
  run_gat(h2, Hid,     Wls, asls, adls, bls, 1,  Out, ei, E, N, xl, sbuf, dbuf, mkeys, denom, logits, ls, 0, stream);

  (void)n_in; (void)out_size; (void)ws_size;
}